// TensorfVM_14482629722460
// MI455X (gfx1250) — compile-verified
//
#include <hip/hip_runtime.h>
#include <cstdint>

// TensoRF VM sampling for MI455X (gfx1250).
// N=512 grid, R=32 components, 3 planes, B*S=524288 samples.
// Gather-bound: matrices (96 MiB) are L2-resident; output (192 MiB) streamed
// with non-temporal stores; per-block coords staged via CDNA5 async
// global->LDS; x-corner pairs fetched as single b64 loads (halves L2 gather
// request count vs 4x b32).

#define NDIM   512
#define RCOMP  32
#define WAVES_PER_BLOCK   8
#define ITERS_PER_WAVE    8
#define SAMPLES_PER_BLOCK (WAVES_PER_BLOCK * ITERS_PER_WAVE)   // 64

// 2-float vector with only 4-byte alignment guarantee (x0 may be odd);
// gfx1250 DWORD alignment mode permits b64 loads at 4-byte alignment.
typedef float f2 __attribute__((ext_vector_type(2), aligned(4)));

__global__ __launch_bounds__(256)
void tensorf_vm_kernel(const float* __restrict__ coords,    // (B*S, 3)
                       const float* __restrict__ matrices,  // (3, R, 512, 512)
                       const float* __restrict__ vectors,   // (3, R, 512, 1)
                       float* __restrict__ out) {           // (B*S, 96)
    __shared__ float sCoords[SAMPLES_PER_BLOCK * 3];        // 768 B slab

    const int tid  = threadIdx.x;
    const int lane = tid & 31;   // = component r
    const int wave = tid >> 5;

    // ---- async-stage this block's contiguous 768B coord slab into LDS ----
    // 6 waves x 32 lanes x 4B = 768B, one async_to_lds issue per wave.
    if (wave < 6) {
        const float* gsrc = coords + (size_t)blockIdx.x * (SAMPLES_PER_BLOCK * 3)
                                   + wave * 32 + lane;
        unsigned            ldsOff = (unsigned)(uintptr_t)(&sCoords[wave * 32 + lane]);
        unsigned long long  gaddr  = (unsigned long long)(uintptr_t)gsrc;
        asm volatile("global_load_async_to_lds_b32 %0, %1, off"
                     :: "v"(ldsOff), "v"(gaddr) : "memory");
        asm volatile("s_wait_asynccnt 0x0" ::: "memory");
    }
    __syncthreads();

    // ---- per-lane vector constants ----
    // Reference's W=1 "bilerp" degenerates: ix==0 always, iy==255.5 always,
    // x1 corner OOB -> vc[i,r] = 0.5*(v[255]+v[256]), constant per (i,r).
    float vc[3];
    #pragma unroll
    for (int i = 0; i < 3; ++i) {
        const float* v = vectors + (size_t)(i * RCOMP + lane) * NDIM;
        vc[i] = 0.5f * (v[255] + v[256]);
    }

    const size_t blockSample0 = (size_t)blockIdx.x * SAMPLES_PER_BLOCK;

    #pragma unroll 2
    for (int it = 0; it < ITERS_PER_WAVE; ++it) {
        const int    sLocal = wave * ITERS_PER_WAVE + it;
        const size_t sample = blockSample0 + (size_t)sLocal;

        // lane-uniform LDS broadcast reads
        const float c0 = sCoords[sLocal * 3 + 0];
        const float c1 = sCoords[sLocal * 3 + 1];
        const float c2 = sCoords[sLocal * 3 + 2];
        const float cc[3] = {c0, c1, c2};

        float* outp = out + sample * (3 * RCOMP) + lane;

        #pragma unroll
        for (int i = 0; i < 3; ++i) {
            // plane i samples at (x, y) = (c[(i+1)%3], c[(i+2)%3])
            const float x = cc[(i + 1) % 3];
            const float y = cc[(i + 2) % 3];

            const float ixf = (x + 1.0f) * 0.5f * (float)(NDIM - 1);
            const float iyf = (y + 1.0f) * 0.5f * (float)(NDIM - 1);
            const float fx0 = floorf(ixf);
            const float fy0 = floorf(iyf);
            const float wx1 = ixf - fx0, wx0 = 1.0f - wx1;
            const float wy1 = iyf - fy0, wy0 = 1.0f - wy1;
            const int x0 = (int)fx0, y0 = (int)fy0;
            const int x1 = x0 + 1,   y1 = y0 + 1;

            // validity (zeros padding) per corner
            const bool vx0 = (unsigned)x0 <= (unsigned)(NDIM - 1);
            const bool vx1 = (unsigned)x1 <= (unsigned)(NDIM - 1);
            const bool vy0 = (unsigned)y0 <= (unsigned)(NDIM - 1);
            const bool vy1 = (unsigned)y1 <= (unsigned)(NDIM - 1);

            // one b64 per row covers both x-corners:
            //   p = { img[yc][xb], img[yc][xb+1] }, xb = clamp(x0, 0, 510)
            const int xb  = min(max(x0, 0), NDIM - 2);
            const int yc0 = min(max(y0, 0), NDIM - 1);
            const int yc1 = min(max(y1, 0), NDIM - 1);

            // this lane's component image: 1 MiB stride, L2-resident
            const float* img = matrices + ((size_t)(i * RCOMP + lane) << 18);
            const f2 p0 = *(const f2*)(img + yc0 * NDIM + xb);
            const f2 p1 = *(const f2*)(img + yc1 * NDIM + xb);

            // select packed texel matching reference's clamp(x) semantics:
            //   corner x0: clamp(x0)==xb   unless x0>510 (then xb+1)
            //   corner x1: clamp(x1)==xb+1 unless x1<1   (then xb)
            const bool hi0 = (x0 > NDIM - 2);
            const bool lo1 = (x1 < 1);
            const float t00 = hi0 ? p0.y : p0.x;
            const float t01 = lo1 ? p0.x : p0.y;
            const float t10 = hi0 ? p1.y : p1.x;
            const float t11 = lo1 ? p1.x : p1.y;

            const float v00 = (vx0 & vy0) ? t00 : 0.0f;
            const float v01 = (vx1 & vy0) ? t01 : 0.0f;
            const float v10 = (vx0 & vy1) ? t10 : 0.0f;
            const float v11 = (vx1 & vy1) ? t11 : 0.0f;

            const float r = v00 * (wy0 * wx0) + v01 * (wy0 * wx1)
                          + v10 * (wy1 * wx0) + v11 * (wy1 * wx1);

            // streaming output: non-temporal so the 192 MiB result doesn't
            // evict the L2-resident matrices
            __builtin_nontemporal_store(r * vc[i], outp + i * RCOMP);
        }
    }
}

extern "C" void kernel_launch(void* const* d_in, const int* in_sizes, int n_in,
                              void* d_out, int out_size, void* d_ws, size_t ws_size,
                              hipStream_t stream) {
    const float* coords   = (const float*)d_in[0];   // (B, S, 3) f32
    const float* matrices = (const float*)d_in[1];   // (3, R, 512, 512) f32
    const float* vectors  = (const float*)d_in[2];   // (3, R, 512, 1) f32
    float*       out      = (float*)d_out;           // (B, S, 96) f32

    const int numSamples = in_sizes[0] / 3;                    // B*S = 524288
    const int blocks     = numSamples / SAMPLES_PER_BLOCK;     // 8192

    tensorf_vm_kernel<<<dim3(blocks), dim3(256), 0, stream>>>(
        coords, matrices, vectors, out);
}